// GraphAttentionLayer_57526791963083
// MI455X (gfx1250) — compile-verified
//
#include <hip/hip_runtime.h>

typedef __attribute__((ext_vector_type(2))) float v2f;
typedef __attribute__((ext_vector_type(8))) float v8f;

#define IN_DIM   64
#define OUT_DIM  64
#define EDGE_DIM 16
#define NNODE    33
#define KNEI     32
#define GB       4                 // batch rows per workgroup
#define ROWS     (GB * NNODE)      // 132
#define MT       9                 // ceil(132/16) M-tiles
#define PADROWS  (MT * 16)         // 144
#define LSTRIDE  68                // floats; 16B-aligned rows -> b128 staging, conflict-free A b64
#define EPS_LN   1e-5f

__launch_bounds__(256)
__global__ void gat_fused_kernel(const float* __restrict__ nodes,
                                 const float* __restrict__ edges,
                                 const float* __restrict__ Ww,
                                 const float* __restrict__ Wb,
                                 const float* __restrict__ Wew,
                                 const float* __restrict__ Web,
                                 const float* __restrict__ attw,
                                 const float* __restrict__ attb,
                                 const float* __restrict__ gamma,
                                 const float* __restrict__ beta,
                                 float* __restrict__ out)
{
    __shared__ float sH[PADROWS * LSTRIDE];   // nodes, overwritten in place by h
    __shared__ float sWt[64 * LSTRIDE];       // W^T : sWt[out_c][in_k]
    __shared__ float sWb[64];
    __shared__ float sAw[192];                // attn_w: [a_t | a_n | a_e]
    __shared__ float sVe[16];                 // We_w @ a_e (folded edge MLP)
    __shared__ float sC[1];                   // We_b . a_e + attn_b
    __shared__ float sAttn[GB * KNEI];
    __shared__ float sGamma[64];
    __shared__ float sBeta[64];

    const int tid  = threadIdx.x;
    const int lane = tid & 31;
    const int wave = tid >> 5;
    const int b0   = blockIdx.x * GB;

    // ------------- Phase 0: stage inputs into LDS (b128 path) -------------
    {
        const float4* g4 = (const float4*)(nodes + (size_t)b0 * NNODE * IN_DIM);
        float4* s4 = (float4*)sH;                       // row r = float4 index r*17
        for (int j = tid; j < ROWS * (IN_DIM / 4); j += 256)
            s4[(j >> 4) * (LSTRIDE / 4) + (j & 15)] = g4[j];
        const float4 z = make_float4(0.f, 0.f, 0.f, 0.f);
        for (int j = tid; j < (PADROWS - ROWS) * (LSTRIDE / 4); j += 256)
            s4[ROWS * (LSTRIDE / 4) + j] = z;           // zero M-padding rows
    }
    for (int i = tid; i < 64 * 64; i += 256) {
        const int k = i >> 6, c = i & 63;
        sWt[c * LSTRIDE + k] = Ww[i];                   // transpose on the fly
    }
    if (tid < 64)  { sWb[tid] = Wb[tid]; sGamma[tid] = gamma[tid]; sBeta[tid] = beta[tid]; }
    if (tid < 192) sAw[tid] = attw[tid];
    if (tid < 16) {                                     // v_e = We_w @ a_e
        float v = 0.f;
        #pragma unroll 8
        for (int o = 0; o < 64; ++o) v += Wew[tid * 64 + o] * attw[128 + o];
        sVe[tid] = v;
    }
    if (tid == 0) {                                     // c = We_b . a_e + attn_b
        float c = attb[0];
        for (int o = 0; o < 64; ++o) c += Web[o] * attw[128 + o];
        sC[0] = c;
    }
    __syncthreads();

    // ------------- Phase 1: h = nodes @ W + b  (V_WMMA_F32_16X16X4_F32) -------------
    {
        const int l15 = lane & 15;
        const int hi  = lane >> 4;     // 0: K={0,1}, 1: K={2,3} per 16x4 f32 A layout
        for (int mt = wave; mt < MT; mt += 8) {
            const int m0 = mt * 16;
            const float* aB = &sH[(m0 + l15) * LSTRIDE + 2 * hi];
            const float* bB = &sWt[l15 * LSTRIDE + 2 * hi];
            v8f a0 = {}, a1 = {}, a2 = {}, a3 = {};
            #pragma unroll
            for (int k0 = 0; k0 < 64; k0 += 4) {
                const v2f af  = *(const v2f*)(aB + k0);
                const v2f b0f = *(const v2f*)(bB + k0);
                const v2f b1f = *(const v2f*)(bB + 16 * LSTRIDE + k0);
                const v2f b2f = *(const v2f*)(bB + 32 * LSTRIDE + k0);
                const v2f b3f = *(const v2f*)(bB + 48 * LSTRIDE + k0);
                a0 = __builtin_amdgcn_wmma_f32_16x16x4_f32(false, af, false, b0f, (short)0, a0, false, false);
                a1 = __builtin_amdgcn_wmma_f32_16x16x4_f32(false, af, false, b1f, (short)0, a1, false, false);
                a2 = __builtin_amdgcn_wmma_f32_16x16x4_f32(false, af, false, b2f, (short)0, a2, false, false);
                a3 = __builtin_amdgcn_wmma_f32_16x16x4_f32(false, af, false, b3f, (short)0, a3, false, false);
            }
            // This wave is the only reader of A-rows [m0, m0+16) -> safe in-place store.
            const float bb0 = sWb[l15], bb1 = sWb[16 + l15], bb2 = sWb[32 + l15], bb3 = sWb[48 + l15];
            const int rb = m0 + 8 * hi;
            #pragma unroll
            for (int v = 0; v < 8; ++v) {
                float* d = &sH[(rb + v) * LSTRIDE + l15];
                d[0]  = a0[v] + bb0;
                d[16] = a1[v] + bb1;
                d[32] = a2[v] + bb2;
                d[48] = a3[v] + bb3;
            }
        }
    }
    __syncthreads();

    // ------------- Phase 2: attention scores (edge GEMM folded into sVe/sC) -------------
    if (tid < GB * KNEI) {
        const int g = tid >> 5;
        const int k = tid & 31;
        const float* hT = &sH[(g * NNODE) * LSTRIDE];
        const float* hN = &sH[(g * NNODE + 1 + k) * LSTRIDE];
        float s = sC[0];
        #pragma unroll 8
        for (int d = 0; d < 64; ++d)
            s += sAw[d] * hT[d] + sAw[64 + d] * hN[d];
        const float4* eg = (const float4*)(edges + ((size_t)(b0 + g) * KNEI + k) * EDGE_DIM);
        #pragma unroll
        for (int e4 = 0; e4 < 4; ++e4) {
            const float4 ev = eg[e4];
            s += sVe[e4 * 4 + 0] * ev.x + sVe[e4 * 4 + 1] * ev.y
               + sVe[e4 * 4 + 2] * ev.z + sVe[e4 * 4 + 3] * ev.w;
        }
        sAttn[tid] = s;
    }
    __syncthreads();

    // ------------- Phase 3: softmax over 32 neighbors (tiny; 1 thread / batch row) -------------
    if (tid < GB) {
        float* sc = &sAttn[tid * KNEI];
        float mx = sc[0];
        for (int k = 1; k < KNEI; ++k) mx = fmaxf(mx, sc[k]);
        float sum = 0.f;
        for (int k = 0; k < KNEI; ++k) { const float e = __expf(sc[k] - mx); sc[k] = e; sum += e; }
        const float inv = 1.0f / sum;
        for (int k = 0; k < KNEI; ++k) sc[k] *= inv;
    }
    __syncthreads();

    // ------------- Phase 4: messages, added into target row -------------
    {
        const int g = tid >> 6;
        const int d = tid & 63;
        float m = 0.f;
        #pragma unroll 8
        for (int k = 0; k < KNEI; ++k)
            m += sAttn[g * KNEI + k] * sH[(g * NNODE + 1 + k) * LSTRIDE + d];
        sH[(g * NNODE) * LSTRIDE + d] += m;
    }
    __syncthreads();

    // ------------- Phase 5: LayerNorm per row (one wave32 per row), b64 stores -------------
    for (int r = wave; r < ROWS; r += 8) {
        const v2f x  = *(const v2f*)&sH[r * LSTRIDE + 2 * lane];   // dims 2l, 2l+1
        float s  = x.x + x.y;
        float ss = x.x * x.x + x.y * x.y;
        #pragma unroll
        for (int off = 16; off >= 1; off >>= 1) {
            s  += __shfl_xor(s,  off, 32);
            ss += __shfl_xor(ss, off, 32);
        }
        const float mu  = s * (1.0f / 64.0f);
        const float var = ss * (1.0f / 64.0f) - mu * mu;
        const float inv = rsqrtf(var + EPS_LN);
        v2f y;
        y.x = (x.x - mu) * inv * sGamma[2 * lane]     + sBeta[2 * lane];
        y.y = (x.y - mu) * inv * sGamma[2 * lane + 1] + sBeta[2 * lane + 1];
        *(v2f*)(out + ((size_t)b0 * NNODE + r) * OUT_DIM + 2 * lane) = y;
    }
}

extern "C" void kernel_launch(void* const* d_in, const int* in_sizes, int n_in,
                              void* d_out, int out_size, void* d_ws, size_t ws_size,
                              hipStream_t stream) {
    const float* nodes = (const float*)d_in[0];
    const float* edges = (const float*)d_in[1];
    const float* Ww    = (const float*)d_in[2];
    const float* Wb    = (const float*)d_in[3];
    const float* Wew   = (const float*)d_in[4];
    const float* Web   = (const float*)d_in[5];
    const float* attw  = (const float*)d_in[6];
    const float* attb  = (const float*)d_in[7];
    const float* gam   = (const float*)d_in[8];
    const float* bet   = (const float*)d_in[9];
    float* out = (float*)d_out;

    const int B = in_sizes[0] / (NNODE * IN_DIM);   // 16384; divisible by GB=4
    const int grid = B / GB;
    gat_fused_kernel<<<grid, 256, 0, stream>>>(nodes, edges, Ww, Wb, Wew, Web,
                                               attw, attb, gam, bet, out);
}